// Net_687194767843
// MI455X (gfx1250) — compile-verified
//
#include <hip/hip_runtime.h>
#include <hip/hip_bf16.h>
#include <math.h>

typedef __attribute__((ext_vector_type(2))) float v2f;
typedef __attribute__((ext_vector_type(8))) float v8f;

#define BG    128      // graphs
#define NPER0 400      // nodes per graph at input
#define HDIM  128      // hidden width
#define NEDGE 819200   // total edges (masked, length constant)
#define KP1   320
#define KP2   256
#define KP3   205

#define KCHUNK   64
#define LDSTRIDE 68    // 68 = 64 + 4 pad: 16B-aligned rows, conflict-free b64 reads

// ---------------------------------------------------------------------------
// Fused GraphConv dual GEMM:
//   Yrel[M,128] = A[M,Kd] @ Wr[128,Kd]^T
//   H   [M,128] = A[M,Kd] @ Wo[128,Kd]^T + bias
// One wave computes 16 M-rows x full 128 N for BOTH outputs:
//   16 v8f accumulators, A fragment reused by 16 WMMAs per K-step.
// Weights staged in LDS in K-chunks of 64 (both matrices, 68KB, 4 blocks/WGP).
// Requires: M % 64 == 0 (exact grid, EXEC all-ones), Kd % 16 == 0.
// ---------------------------------------------------------------------------
__global__ void __launch_bounds__(128)
gconv_dual_gemm(const float* __restrict__ A,
                const float* __restrict__ Wr,
                const float* __restrict__ Wo,
                const float* __restrict__ bias,
                float* __restrict__ Yrel,
                float* __restrict__ H,
                int M, int Kd)
{
    __shared__ float lr[HDIM * LDSTRIDE];
    __shared__ float lo[HDIM * LDSTRIDE];

    int lane  = threadIdx.x & 31;
    int wave  = threadIdx.x >> 5;
    int tileM = blockIdx.x * 4 + wave;

    int r  = lane & 15;          // A: local row  / B,C: local col
    int kq = (lane >> 4) * 2;    // K pair select (0 or 2)
    const float* a_row = A + (size_t)(tileM * 16 + r) * Kd;

    v8f accR[8] = {};
    v8f accH[8] = {};

    for (int kc = 0; kc < Kd; kc += KCHUNK) {
        int kcs = (Kd - kc < KCHUNK) ? (Kd - kc) : KCHUNK;
        // --- cooperative stage: thread t <-> weight row t (128 rows) ---
        {
            int t = threadIdx.x;
            const float4* gr = (const float4*)(Wr + (size_t)t * Kd + kc);
            const float4* go = (const float4*)(Wo + (size_t)t * Kd + kc);
            float4* sr = (float4*)(lr + t * LDSTRIDE);
            float4* so = (float4*)(lo + t * LDSTRIDE);
            int nq = kcs >> 2;
            for (int q = 0; q < nq; ++q) { sr[q] = gr[q]; so[q] = go[q]; }
            if (kc + KCHUNK < Kd) {   // prefetch next chunk -> global_prefetch_b8
                __builtin_prefetch(Wr + (size_t)t * Kd + kc + KCHUNK, 0, 3);
                __builtin_prefetch(Wo + (size_t)t * Kd + kc + KCHUNK, 0, 3);
            }
        }
        __syncthreads();
        // --- compute: 16 WMMAs per K-step, A fragment loaded once ---
        for (int kk = 0; kk < kcs; kk += 4) {
            v2f a = *(const v2f*)(a_row + kc + kk + kq);
            #pragma unroll
            for (int n = 0; n < 8; ++n) {
                v2f br = *(const v2f*)(lr + (n * 16 + r) * LDSTRIDE + kk + kq);
                v2f bo = *(const v2f*)(lo + (n * 16 + r) * LDSTRIDE + kk + kq);
                accR[n] = __builtin_amdgcn_wmma_f32_16x16x4_f32(false, a, false, br,
                                                                (short)0, accR[n], false, false);
                accH[n] = __builtin_amdgcn_wmma_f32_16x16x4_f32(false, a, false, bo,
                                                                (short)0, accH[n], false, false);
            }
        }
        __syncthreads();
    }

    int rbase = tileM * 16 + (lane >> 4) * 8;
    #pragma unroll
    for (int n = 0; n < 8; ++n) {
        int colg = n * 16 + r;
        float bv = bias[colg];
        #pragma unroll
        for (int vr = 0; vr < 8; ++vr) {
            Yrel[(size_t)(rbase + vr) * HDIM + colg] = accR[n][vr];
            H   [(size_t)(rbase + vr) * HDIM + colg] = accH[n][vr] + bv;
        }
    }
}

// ---------------------------------------------------------------------------
// Generic WMMA f32 GEMM (MLP head):  Y = A[M,Kd] @ W[N,Kd]^T (+bias)(+relu)
// ---------------------------------------------------------------------------
__global__ void wmma_gemm_f32(const float* __restrict__ A,
                              const float* __restrict__ W,
                              const float* __restrict__ bias,
                              float* __restrict__ Y,
                              int M, int Kd, int N, int relu)
{
    int lane  = threadIdx.x & 31;
    int wave  = threadIdx.x >> 5;
    int tileN = blockIdx.y;
    int tileM = blockIdx.x * 4 + wave;
    if (tileM * 16 >= M) return;   // wave-uniform

    int r  = lane & 15;
    int kq = (lane >> 4) * 2;
    const float* a_ptr = A + (size_t)(tileM * 16 + r) * Kd + kq;
    const float* b_ptr = W + (size_t)(tileN * 16 + r) * Kd + kq;

    v8f c = {};
    for (int k0 = 0; k0 < Kd; k0 += 4) {
        v2f a = *(const v2f*)(a_ptr + k0);
        v2f b = *(const v2f*)(b_ptr + k0);
        c = __builtin_amdgcn_wmma_f32_16x16x4_f32(false, a, false, b,
                                                  (short)0, c, false, false);
    }
    int colg  = tileN * 16 + r;
    float bv  = bias ? bias[colg] : 0.0f;
    int rbase = tileM * 16 + (lane >> 4) * 8;
    #pragma unroll
    for (int vr = 0; vr < 8; ++vr) {
        float v = c[vr] + bv;
        if (relu) v = fmaxf(v, 0.0f);
        Y[(size_t)(rbase + vr) * N + colg] = v;
    }
}

// ---------------------------------------------------------------------------
// Edge aggregation: h[dst] += yrel[src] * em  (128 feats, 32 threads/edge)
// ---------------------------------------------------------------------------
__global__ void edge_scatter(const float* __restrict__ ysrc,
                             const int* __restrict__ src,
                             const int* __restrict__ dst,
                             const float* __restrict__ em,   // null => 1.0
                             float* __restrict__ hout, int nE)
{
    long long t = (long long)blockIdx.x * blockDim.x + threadIdx.x;
    int e  = (int)(t >> 5);
    int f0 = (int)(t & 31);
    if (e >= nE) return;
    float m = em ? em[e] : 1.0f;
    if (m == 0.0f) return;
    const float* yp = ysrc + (size_t)src[e] * HDIM;
    float*       hp = hout + (size_t)dst[e] * HDIM;
    #pragma unroll
    for (int i = 0; i < 4; ++i) {
        int f = f0 + i * 32;
        atomicAdd(hp + f, yp[f] * m);
    }
}

__global__ void relu_kernel(float* __restrict__ h, long long n)
{
    long long t = (long long)blockIdx.x * blockDim.x + threadIdx.x;
    if (t < n) h[t] = fmaxf(h[t], 0.0f);
}

// 1 / (||w||_2 + 1e-16), w has HDIM=128 elems
__global__ void invnorm_kernel(const float* __restrict__ w, float* __restrict__ out)
{
    __shared__ float red[HDIM];
    int t = threadIdx.x;
    float v = w[t];
    red[t] = v * v;
    __syncthreads();
    for (int s = 64; s > 0; s >>= 1) {
        if (t < s) red[t] += red[t + s];
        __syncthreads();
    }
    if (t == 0) out[0] = 1.0f / (sqrtf(red[0]) + 1e-16f);
}

// score[node] = dot(h[node], w) * invnorm  — one wave per node
__global__ void score_kernel(const float* __restrict__ h,
                             const float* __restrict__ w,
                             const float* __restrict__ invn,
                             float* __restrict__ score, int Nn)
{
    int lane = threadIdx.x & 31;
    int node = (int)(((long long)blockIdx.x * blockDim.x + threadIdx.x) >> 5);
    if (node >= Nn) return;
    const float* hp = h + (size_t)node * HDIM;
    float acc = 0.0f;
    #pragma unroll
    for (int i = 0; i < 4; ++i) acc += hp[lane + i * 32] * w[lane + i * 32];
    for (int off = 16; off > 0; off >>= 1) acc += __shfl_down(acc, off, 32);
    if (lane == 0) score[node] = acc * invn[0];
}

// ---------------------------------------------------------------------------
// Per-graph top-k via 512-wide bitonic sort in LDS (desc value, asc index).
// Intra-graph ordering of the selected nodes provably does not affect the
// network output (readout = max/mean, aggregation = sum): only set + gates.
// ---------------------------------------------------------------------------
__global__ void topk_kernel(const float* __restrict__ score, int npg, int k,
                            int* __restrict__ sel_idx, float* __restrict__ sel_val)
{
    __shared__ float sv[512];
    __shared__ int   si[512];
    int g = blockIdx.x;
    for (int i = threadIdx.x; i < 512; i += blockDim.x) {
        if (i < npg) { sv[i] = score[(size_t)g * npg + i]; si[i] = i; }
        else         { sv[i] = -3.402823466e38f;           si[i] = i; }
    }
    __syncthreads();
    for (int ks = 2; ks <= 512; ks <<= 1) {
        for (int j = ks >> 1; j > 0; j >>= 1) {
            for (int i = threadIdx.x; i < 512; i += blockDim.x) {
                int ixj = i ^ j;
                if (ixj > i) {
                    float va = sv[i], vb = sv[ixj];
                    int   ia = si[i], ib = si[ixj];
                    bool a_first = (va > vb) || (va == vb && ia < ib);
                    bool desc = ((i & ks) == 0);
                    bool sw = desc ? !a_first : a_first;
                    if (sw) { sv[i] = vb; sv[ixj] = va; si[i] = ib; si[ixj] = ia; }
                }
            }
            __syncthreads();
        }
    }
    for (int t = threadIdx.x; t < k; t += blockDim.x) {
        sel_idx[(size_t)g * k + t] = si[t];
        sel_val[(size_t)g * k + t] = sv[t];
    }
}

// gather + tanh gate + new_id assignment (new_id pre-filled with -1)
__global__ void gather_gate_kernel(const float* __restrict__ h,
                                   const int* __restrict__ sel_idx,
                                   const float* __restrict__ sel_val,
                                   float* __restrict__ hk,
                                   int* __restrict__ newid, int k, int npg)
{
    long long t = (long long)blockIdx.x * blockDim.x + threadIdx.x;
    long long row = t >> 7;
    int f = (int)(t & 127);
    if (row >= (long long)BG * k) return;
    int g = (int)(row / k);
    int li = sel_idx[row];
    float gate = tanhf(sel_val[row]);
    hk[row * HDIM + f] = h[((size_t)g * npg + li) * HDIM + f] * gate;
    if (f == 0) newid[(size_t)g * npg + li] = (int)row;
}

__global__ void remap_kernel(const int* __restrict__ si, const int* __restrict__ di,
                             const float* __restrict__ emi,          // null => 1.0
                             const int* __restrict__ newid,
                             int* __restrict__ so, int* __restrict__ do_,
                             float* __restrict__ emo, int nE)
{
    int e = blockIdx.x * blockDim.x + threadIdx.x;
    if (e >= nE) return;
    int s = si[e], d = di[e];
    float m = emi ? emi[e] : 1.0f;
    int ns = newid[s], nd = newid[d];
    bool keep = (ns >= 0) && (nd >= 0) && (m != 0.0f);
    so[e]  = keep ? ns : 0;
    do_[e] = keep ? nd : 0;
    emo[e] = keep ? m : 0.0f;
}

// accumulate [max ; mean] readout into zsum[BG,256]
__global__ void readout_kernel(const float* __restrict__ hk, int k,
                               float* __restrict__ zsum)
{
    int g = blockIdx.x, f = threadIdx.x; // 128 threads
    const float* hp = hk + ((size_t)g * k) * HDIM + f;
    float mx = -3.402823466e38f, sm = 0.0f;
    for (int r = 0; r < k; ++r) {
        float v = hp[(size_t)r * HDIM];
        mx = fmaxf(mx, v);
        sm += v;
    }
    zsum[(size_t)g * 256 + f]       += mx;
    zsum[(size_t)g * 256 + 128 + f] += sm / (float)k;
}

// final classifier: logits[b] = z2[b]@W3^T + bl3 ; log_softmax over C=2
__global__ void final_kernel(const float* __restrict__ z2,
                             const float* __restrict__ W3,
                             const float* __restrict__ bl3,
                             float* __restrict__ out)
{
    int b = blockIdx.x * blockDim.x + threadIdx.x;
    if (b >= BG) return;
    const float* zp = z2 + (size_t)b * 64;
    float v0 = bl3[0], v1 = bl3[1];
    for (int i = 0; i < 64; ++i) {
        float z = zp[i];
        v0 += z * W3[i];
        v1 += z * W3[64 + i];
    }
    float m = fmaxf(v0, v1);
    float lse = m + logf(expf(v0 - m) + expf(v1 - m));
    out[b * 2 + 0] = v0 - lse;
    out[b * 2 + 1] = v1 - lse;
}

// ---------------------------------------------------------------------------
extern "C" void kernel_launch(void* const* d_in, const int* in_sizes, int n_in,
                              void* d_out, int out_size, void* d_ws, size_t ws_size,
                              hipStream_t stream)
{
    (void)in_sizes; (void)n_in; (void)out_size; (void)ws_size;
    const float* x     = (const float*)d_in[0];
    const int*   esrc  = (const int*)  d_in[1];
    const int*   edst  = (const int*)  d_in[2];
    const float* Wrel[3]  = { (const float*)d_in[4],  (const float*)d_in[8],  (const float*)d_in[12] };
    const float* Wroot[3] = { (const float*)d_in[5],  (const float*)d_in[9],  (const float*)d_in[13] };
    const float* bvec[3]  = { (const float*)d_in[6],  (const float*)d_in[10], (const float*)d_in[14] };
    const float* pw[3]    = { (const float*)d_in[7],  (const float*)d_in[11], (const float*)d_in[15] };
    const float* W1  = (const float*)d_in[16]; const float* bl1 = (const float*)d_in[17];
    const float* W2  = (const float*)d_in[18]; const float* bl2 = (const float*)d_in[19];
    const float* W3  = (const float*)d_in[20]; const float* bl3 = (const float*)d_in[21];
    float* out = (float*)d_out;

    char* ws = (char*)d_ws;
    float* yrel   = (float*)(ws + 0);            // 51200*128 f32
    float* h      = (float*)(ws + 26214400);     // 51200*128 f32
    float* hk     = (float*)(ws + 52428800);     // 40960*128 f32 (max)
    float* score  = (float*)(ws + 73400320);     // 51200 f32
    int*   selidx = (int*)  (ws + 73605120);     // 128*320
    float* selval = (float*)(ws + 73768960);     // 128*320
    int*   newid  = (int*)  (ws + 73932800);     // 51200
    int*   eA_s   = (int*)  (ws + 74137600);     // E
    int*   eA_d   = (int*)  (ws + 77414400);
    float* eA_m   = (float*)(ws + 80691200);
    int*   eB_s   = (int*)  (ws + 83968000);
    int*   eB_d   = (int*)  (ws + 87244800);
    float* eB_m   = (float*)(ws + 90521600);
    float* zsum   = (float*)(ws + 93798400);     // 128*256
    float* z1     = (float*)(ws + 93929472);     // 128*128
    float* z2     = (float*)(ws + 93995008);     // 128*64
    float* invn   = (float*)(ws + 94027776);     // 1

    hipMemsetAsync(zsum, 0, (size_t)BG * 256 * sizeof(float), stream);

    const int npg[3]  = { NPER0, KP1, KP2 };
    const int kpp[3]  = { KP1,  KP2, KP3 };
    const int fin[3]  = { 400, HDIM, HDIM };

    const float* Xin = x;
    const int *cs = esrc, *cd = edst; const float* cm = nullptr;
    int*   ns_[2] = { eA_s, eB_s };
    int*   nd_[2] = { eA_d, eB_d };
    float* nm_[2] = { eA_m, eB_m };

    for (int L = 0; L < 3; ++L) {
        int Nn = BG * npg[L];
        int Kd = fin[L];
        // fused: yrel = Xin@Wrel^T ; h = Xin@Wroot^T + b   (Xin read once)
        gconv_dual_gemm<<<Nn / 64, 128, 0, stream>>>(Xin, Wrel[L], Wroot[L], bvec[L],
                                                     yrel, h, Nn, Kd);
        // h[dst] += yrel[src] * em   (aggregate post-GEMM: 128-wide, not 400)
        long long sthreads = (long long)NEDGE * 32;
        edge_scatter<<<(int)((sthreads + 255) / 256), 256, 0, stream>>>(yrel, cs, cd, cm, h, NEDGE);
        long long hn = (long long)Nn * HDIM;
        relu_kernel<<<(int)((hn + 255) / 256), 256, 0, stream>>>(h, hn);
        // top-k pool
        invnorm_kernel<<<1, HDIM, 0, stream>>>(pw[L], invn);
        score_kernel<<<(Nn * 32 + 255) / 256, 256, 0, stream>>>(h, pw[L], invn, score, Nn);
        topk_kernel<<<BG, 256, 0, stream>>>(score, npg[L], kpp[L], selidx, selval);
        hipMemsetAsync(newid, 0xFF, (size_t)Nn * sizeof(int), stream); // -1
        long long grows = (long long)BG * kpp[L] * HDIM;
        gather_gate_kernel<<<(int)((grows + 255) / 256), 256, 0, stream>>>(
            h, selidx, selval, hk, newid, kpp[L], npg[L]);
        if (L < 2) { // edges after the last pool are dead code
            remap_kernel<<<(NEDGE + 255) / 256, 256, 0, stream>>>(
                cs, cd, cm, newid, ns_[L], nd_[L], nm_[L], NEDGE);
            cs = ns_[L]; cd = nd_[L]; cm = nm_[L];
        }
        readout_kernel<<<BG, HDIM, 0, stream>>>(hk, kpp[L], zsum);
        Xin = hk;
    }

    // MLP head: z1 = relu(zsum@W1^T+bl1) ; z2 = relu(z1@W2^T+bl2) ; log_softmax(z2@W3^T+bl3)
    { dim3 g1(BG / 64, 128 / 16); wmma_gemm_f32<<<g1, 128, 0, stream>>>(zsum, W1, bl1, z1, BG, 256, 128, 1); }
    { dim3 g2(BG / 64, 64 / 16);  wmma_gemm_f32<<<g2, 128, 0, stream>>>(z1,   W2, bl2, z2, BG, 128, 64, 1); }
    final_kernel<<<1, BG, 0, stream>>>(z2, W3, bl3, out);
}